// PEPS_delocalized_Model_3178275799241
// MI455X (gfx1250) — compile-verified
//
#include <hip/hip_runtime.h>

// CDNA5 (gfx1250) PEPS amplitude kernel, branchless edition.
// Boundary-vector MPO contraction; each site is a 9x9 @ 9x243 GEMM mapped to
// V_WMMA_F32_16X16X4_F32 (M=16 pad, N=16 col tiles, K=4 x 3 chunks).
// All guards are implemented via zero-padded LDS tiles and cndmask address
// selection so no EXEC-mask branching surrounds the ds/wmma pipeline.

typedef __attribute__((ext_vector_type(2))) float v2f;
typedef __attribute__((ext_vector_type(8))) float v8f;

#define LX 6
#define LY 6
#define NB 16
#define V_LEN 729                 // 3^6 boundary vector
#define WSTRIDE 256               // padded column stride (valid 243)
#define WROWS_PAD 12              // padded K rows (valid 9)
#define W_PAD_LEN (WROWS_PAD * WSTRIDE)   // 3072 floats per buffer
#define ASTRIDE 12                // padded A3 row stride (valid 9)
#define AS_LEN (16 * ASTRIDE)     // padded A3 tile: 16 M rows x 12 K cols
#define NTHREADS 128

__global__ __launch_bounds__(NTHREADS)
void peps_amp_kernel(const int* __restrict__ xs_all,
                     const float* __restrict__ T,
                     float* __restrict__ out)
{
    __shared__ float Wbuf[2][W_PAD_LEN];  // ping-pong working tensor (padded)
    __shared__ float As[AS_LEN];          // per-site A3 tile (padded)
    __shared__ float v[V_LEN];            // boundary vector
    __shared__ float dump[1];             // sink for masked-off stores
    __shared__ int   sx[LX * LY];         // this sample's spins

    const int tid    = threadIdx.x;
    const int sample = blockIdx.x;
    const int lane   = tid & 31;          // wave32
    const int waveId = tid >> 5;          // 4 waves
    const int khalf  = lane >> 4;         // K-pair half (ISA 16x4 f32 layout)
    const int lrow   = lane & 15;         // M row (A) / N col (B,D)

    // One-time zeroing: pad regions of Wbuf/As must be zero and stay zero.
    float* Wflat = &Wbuf[0][0];
    for (int i = tid; i < 2 * W_PAD_LEN; i += NTHREADS) Wflat[i] = 0.0f;
    for (int i = tid; i < AS_LEN; i += NTHREADS) As[i] = 0.0f;
    if (tid < LX * LY) sx[tid] = xs_all[sample * LX * LY + tid];
    // v = e_0 : top boundary (all u = 0) delta
    for (int i = tid; i < V_LEN; i += NTHREADS) v[i] = (i == 0) ? 1.0f : 0.0f;
    __syncthreads();

    int cur = 0;
    for (int x = 0; x < LX; ++x) {
        // W_{-1}[(l,u0),(u1..u5)] from v; rows l != 0 are zero (l0 = 0 BC).
        // Branchless: value select, index always in range.
        float* Wc0 = Wbuf[cur];
        for (int i = tid; i < 9 * 243; i += NTHREADS) {
            int kappa = i / 243;
            int c     = i - kappa * 243;
            int l     = kappa / 3;
            int u0    = kappa - l * 3;
            float val = v[u0 * 243 + c];
            Wc0[kappa * WSTRIDE + c] = (l == 0) ? val : 0.0f;
        }
        __syncthreads();

        #pragma unroll
        for (int y = 0; y < LY; ++y) {
            float* Wc = Wbuf[cur];
            float* Wn = Wbuf[cur ^ 1];

            // Spin bits -> physical index selector for this site.
            int p  = sx[x * LY + y];
            int pu = (x > 0)      ? sx[(x - 1) * LY + y] : 0;
            int pd = (x < LX - 1) ? sx[(x + 1) * LY + y] : 0;
            int pl = (y > 0)      ? sx[x * LY + y - 1]   : 0;
            int pr = (y < LY - 1) ? sx[x * LY + y + 1]   : 0;
            int pbits = ((((p * 2 + pu) * 2 + pd) * 2 + pl) * 2 + pr);
            const float* Tb = T + (size_t)(x * LY + y) * 81 * 32 + pbits;

            // Phase A: cooperative gather of A3[(r,d),(l,u)] into padded LDS.
            if (tid < 81) {
                int rho = tid / 9;             // r*3 + d
                int kap = tid - rho * 9;       // l*3 + u
                int r = rho / 3, d = rho - r * 3;
                int l = kap / 3, u = kap - l * 3;
                As[rho * ASTRIDE + kap] = Tb[(u * 27 + d * 9 + l * 3 + r) * 32];
            }
            // Prefetch next site's T slice (L2-resident; gfx1250 prefetch path).
            if (y < LY - 1) __builtin_prefetch(Tb + 81 * 32, 0, 1);
            __syncthreads();

            // A fragments (unconditional padded LDS reads).
            // ISA 16x4 f32 A layout: lanes 0-15 hold K0/K1, lanes 16-31 K2/K3.
            v2f afr[3];
            #pragma unroll
            for (int kc = 0; kc < 3; ++kc) {
                int k0 = kc * 4 + khalf * 2;
                afr[kc].x = As[lrow * ASTRIDE + k0];
                afr[kc].y = As[lrow * ASTRIDE + k0 + 1];
            }

            // Phase B: each wave computes 4 of the 16 column tiles.
            for (int t = 0; t < 4; ++t) {
                const int tl = waveId * 4 + t;
                const int cg = tl * 16 + lrow;      // 0..255, pad cols are zero

                v8f acc = {0.f, 0.f, 0.f, 0.f, 0.f, 0.f, 0.f, 0.f};
                #pragma unroll
                for (int kc = 0; kc < 3; ++kc) {
                    int k0 = kc * 4 + khalf * 2;    // B mirrors A's K split
                    v2f bfr;
                    bfr.x = Wc[ k0      * WSTRIDE + cg];
                    bfr.y = Wc[(k0 + 1) * WSTRIDE + cg];
                    acc = __builtin_amdgcn_wmma_f32_16x16x4_f32(
                        false, afr[kc], false, bfr, (short)0, acc, false, false);
                }

                // Scatter with digit permutation
                //   (r, d_y, dpref, u_next, urest) -> (r, u_next, dpref, d_y, urest)
                // Masked lanes store to the dump word (address select, no branch).
                if (y < LY - 1) {
                    const int pwtab[5] = {81, 27, 9, 3, 1};
                    const int P = pwtab[y];         // 3^(4-y), const after unroll
                    #pragma unroll
                    for (int j = 0; j < 8; ++j) {
                        int m = j + 8 * khalf;      // D row = r*3 + d_y
                        int r = m / 3, dd = m - r * 3;
                        int dpref = cg / (3 * P);
                        int rem   = cg - dpref * 3 * P;
                        int un    = rem / P;
                        int urest = rem - un * P;
                        bool ok = (m < 9) && (cg < 243);
                        float* dst = ok
                            ? &Wn[(r * 3 + un) * WSTRIDE + (dpref * 3 + dd) * P + urest]
                            : &dump[0];
                        *dst = acc[j];
                    }
                } else {
                    // Last site of row: keep r5 == 0, fold into new v[d0..d5].
                    #pragma unroll
                    for (int j = 0; j < 8; ++j) {
                        int m = j + 8 * khalf;      // m = r5*3 + d5
                        bool ok = (m < 3) && (cg < 243);
                        float* dst = ok ? &v[cg * 3 + m] : &dump[0];
                        *dst = acc[j];
                    }
                }
            }
            __syncthreads();
            cur ^= 1;
        }
    }
    // Bottom boundary: all d = 0  ->  amplitude = v[0]
    if (tid == 0) out[sample] = v[0];
}

extern "C" void kernel_launch(void* const* d_in, const int* in_sizes, int n_in,
                              void* d_out, int out_size, void* d_ws, size_t ws_size,
                              hipStream_t stream) {
    const int*   xs = (const int*)d_in[0];    // x: (16, 36) int32
    const float* T  = (const float*)d_in[1];  // T: (6,6,3,3,3,3,2,2,2,2,2) f32
    float* out = (float*)d_out;               // (16,) f32
    (void)in_sizes; (void)n_in; (void)out_size; (void)d_ws; (void)ws_size;
    peps_amp_kernel<<<dim3(NB), dim3(NTHREADS), 0, stream>>>(xs, T, out);
}